// SNN_32865089749055
// MI455X (gfx1250) — compile-verified
//
#include <hip/hip_runtime.h>
#include <hip/hip_bf16.h>
#include <stdint.h>

// ---------------- problem constants ----------------
#define BB    16
#define NN    8192
#define DD    8
#define NEXT  4096
#define RTOT  (NN + NEXT)          // 12288 candidate K rows (internal + external)

#define ALPHA 0.95f
#define RHO   0.99f
#define BETA  1.8f
#define TH0   1.0f

// ---------------- workspace layout (bytes) ----------------
#define HDR_OFF  ((size_t)0)                       // 2 ints: [0]=nchunks, [1]=K'
#define FLG_OFF  ((size_t)256)                     // RTOT ints (active-row flags)
#define ADR_OFF  (FLG_OFF + (size_t)RTOT * 4)      // (RTOT+4) u64 row addresses
#define PRE_OFF  (ADR_OFF + (size_t)(RTOT + 4) * 8)// RTOT*16 f32 staged activations
#define ACM_OFF  (PRE_OFF + (size_t)RTOT * 16 * 4) // compacted A in WMMA lane layout

typedef __attribute__((ext_vector_type(2))) float v2f;
typedef __attribute__((ext_vector_type(8))) float v8f;
// global (addrspace 1) pointer: forces global_load_* instead of flat_load_*
typedef const float __attribute__((address_space(1)))* gfp;

// ===========================================================================
// Kernel 1: elementwise ALIF update + delay gather + activity flags.
// ===========================================================================
__global__ void __launch_bounds__(128) snn_prep(
    const float* __restrict__ V, const float* __restrict__ a,
    const float* __restrict__ Xbuf, const int* __restrict__ dmap,
    const float* __restrict__ Xext,
    float* __restrict__ outX, float* __restrict__ outV, float* __restrict__ outA,
    int* __restrict__ flags, float* __restrict__ preeff)
{
    int row = blockIdx.x * blockDim.x + threadIdx.x;
    if (row >= RTOT) return;

    int any = 0;
    if (row < NN) {
        const int i = row;
        int dsel = 0;
#pragma unroll
        for (int d = 0; d < DD; ++d)
            if (dmap[d * NN + i] != 0) dsel = d;

#pragma unroll
        for (int b = 0; b < BB; ++b) {
            const size_t o = (size_t)b * NN + i;
            float pe = Xbuf[((size_t)dsel * BB + b) * NN + i];
            preeff[(size_t)row * BB + b] = pe;
            any |= (pe != 0.0f);
            float v  = V[o];
            float av = a[o];
            float u  = v - (TH0 + BETA * av);
            float X  = (u >= 0.0f) ? 1.0f : 0.0f;
            outX[o] = X;
            outA[o] = RHO * av + X;
            outV[o] = ALPHA * v * (1.0f - X);   // partial; GEMM adds `current`
        }
    } else {
        const int j = row - NN;
#pragma unroll
        for (int b = 0; b < BB; ++b) {
            float pe = Xext[(size_t)b * NEXT + j];
            preeff[(size_t)row * BB + b] = pe;
            any |= (pe != 0.0f);
        }
    }
    flags[row] = any;
}

// ===========================================================================
// Kernel 2: deterministic compaction of active K rows (single 1024-thread
// block, Hillis-Steele scan).
// ===========================================================================
__global__ void __launch_bounds__(1024) snn_compact(
    const int* __restrict__ flags, const float* __restrict__ preeff,
    const float* __restrict__ Wint, const float* __restrict__ Wext,
    unsigned long long* __restrict__ addrTab, float* __restrict__ acomp,
    int* __restrict__ hdr)
{
    __shared__ int sc[1024];
    const int tid = threadIdx.x;
    int running = 0;

    for (int base = 0; base < RTOT; base += 1024) {
        const int row = base + tid;           // RTOT % 1024 == 0
        const int f = flags[row];
        sc[tid] = f;
        __syncthreads();
        for (int off = 1; off < 1024; off <<= 1) {
            int v = (tid >= off) ? sc[tid - off] : 0;
            __syncthreads();
            sc[tid] += v;
            __syncthreads();
        }
        const int excl = sc[tid] - f;
        if (f) {
            const int p = running + excl;
            addrTab[p] = (row < NN)
                ? (unsigned long long)(uintptr_t)(Wint + (size_t)row * NN)
                : (unsigned long long)(uintptr_t)(Wext + (size_t)(row - NN) * NN);
            const int c = p >> 2, r = p & 3;
            const int lo = (r >= 2) ? 16 : 0;
            const int jj = r & 1;
#pragma unroll
            for (int b = 0; b < BB; ++b)
                acomp[(size_t)c * 64 + (size_t)(lo + b) * 2 + jj] =
                    preeff[(size_t)row * BB + b];
        }
        __syncthreads();
        running += sc[1023];
        __syncthreads();
    }

    const int K    = running;
    const int Kpad = (K + 3) & ~3;
    if (tid < (Kpad - K)) {
        const int k = K + tid;
        addrTab[k] = (unsigned long long)(uintptr_t)Wint;   // safe dummy row
        const int c = k >> 2, r = k & 3;
        const int lo = (r >= 2) ? 16 : 0;
        const int jj = r & 1;
#pragma unroll
        for (int b = 0; b < BB; ++b)
            acomp[(size_t)c * 64 + (size_t)(lo + b) * 2 + jj] = 0.0f;
    }
    if (tid == 0) { hdr[0] = Kpad >> 2; hdr[1] = K; }
}

// ===========================================================================
// Kernel 3: skinny GEMM over compacted K using V_WMMA_F32_16X16X4_F32.
// 256 blocks, each owning TWO 16-column tiles (shared A / row addresses),
// 8-way K-split per block. Wave id is readfirstlane'd so the K loop is
// wave-uniform: EXEC stays all-ones for WMMA and the address-table reads
// scalarize to s_load (constant cache) instead of per-lane vector loads.
// ===========================================================================
__global__ void __launch_bounds__(256) snn_gemm(
    const unsigned long long* __restrict__ addrTab,
    const float* __restrict__ acomp,
    const int* __restrict__ hdr,
    float* __restrict__ outV)
{
    __shared__ float red[8][32][8];

    const int j0   = blockIdx.x * 32;                       // two 16-col tiles
    const int w    = __builtin_amdgcn_readfirstlane(threadIdx.x >> 5);
    const int lane = threadIdx.x & 31;
    const int nlo  = lane & 15;
    const bool hi  = lane >= 16;
    const int nch  = __builtin_amdgcn_readfirstlane(hdr[0]);

    v8f c0 = {};
    v8f c1 = {};
    const float2* __restrict__ acomp2 = (const float2*)acomp;

#pragma unroll 4
    for (int ci = w; ci < nch; ci += 8) {
        // wave-uniform address fetch -> scalar loads
        const unsigned long long a0 = addrTab[4 * ci + 0];
        const unsigned long long a1 = addrTab[4 * ci + 1];
        const unsigned long long a2 = addrTab[4 * ci + 2];
        const unsigned long long a3 = addrTab[4 * ci + 3];

        // A operand: pre-packed WMMA lane layout, one coalesced 8B load/lane
        const float2 av = acomp2[(size_t)ci * 32 + lane];

        // B operand rows: lanes 0-15 -> K=+0/+1, lanes 16-31 -> K=+2/+3
        const gfp r0 = (gfp)(uintptr_t)(hi ? a2 : a0);
        const gfp r1 = (gfp)(uintptr_t)(hi ? a3 : a1);

        const float b00 = __builtin_nontemporal_load(r0 + (j0 + nlo));
        const float b10 = __builtin_nontemporal_load(r1 + (j0 + nlo));
        const float b01 = __builtin_nontemporal_load(r0 + (j0 + 16 + nlo));
        const float b11 = __builtin_nontemporal_load(r1 + (j0 + 16 + nlo));

        v2f A;   A[0]  = av.x; A[1]  = av.y;
        v2f Bv0; Bv0[0] = b00; Bv0[1] = b10;
        v2f Bv1; Bv1[0] = b01; Bv1[1] = b11;

        c0 = __builtin_amdgcn_wmma_f32_16x16x4_f32(
                 false, A, false, Bv0, (short)0, c0, false, false);
        c1 = __builtin_amdgcn_wmma_f32_16x16x4_f32(
                 false, A, false, Bv1, (short)0, c1, false, false);
    }

    // ---- reduce K-split partials and fuse V_new epilogue (tile 0) ----
#pragma unroll
    for (int v = 0; v < 8; ++v) red[w][lane][v] = c0[v];
    __syncthreads();
    if (w == 0) {
#pragma unroll
        for (int v = 0; v < 8; ++v) {
            float s = 0.0f;
#pragma unroll
            for (int ww = 0; ww < 8; ++ww) s += red[ww][lane][v];
            const int b = v + (hi ? 8 : 0);          // C layout: vgpr v -> M=v / M=v+8
            const size_t o = (size_t)b * NN + (j0 + nlo);
            outV[o] = outV[o] + s;                   // reset-partial + current
        }
    }
    __syncthreads();

    // ---- tile 1 ----
#pragma unroll
    for (int v = 0; v < 8; ++v) red[w][lane][v] = c1[v];
    __syncthreads();
    if (w == 0) {
#pragma unroll
        for (int v = 0; v < 8; ++v) {
            float s = 0.0f;
#pragma unroll
            for (int ww = 0; ww < 8; ++ww) s += red[ww][lane][v];
            const int b = v + (hi ? 8 : 0);
            const size_t o = (size_t)b * NN + (j0 + 16 + nlo);
            outV[o] = outV[o] + s;
        }
    }
}

// ===========================================================================
extern "C" void kernel_launch(void* const* d_in, const int* in_sizes, int n_in,
                              void* d_out, int out_size, void* d_ws, size_t ws_size,
                              hipStream_t stream)
{
    const float* V    = (const float*)d_in[0];
    const float* a    = (const float*)d_in[1];
    const float* Xbuf = (const float*)d_in[2];
    const int*   dmap = (const int*)d_in[3];   // bool -> int per harness convention
    const float* Wint = (const float*)d_in[4];
    const float* Xext = (const float*)d_in[5];
    const float* Wext = (const float*)d_in[6];

    float* outX = (float*)d_out;               // [16,8192]
    float* outV = outX + (size_t)BB * NN;      // [16,8192]
    float* outA = outV + (size_t)BB * NN;      // [16,8192]

    char* ws = (char*)d_ws;
    int*                 hdr     = (int*)(ws + HDR_OFF);
    int*                 flags   = (int*)(ws + FLG_OFF);
    unsigned long long*  addrTab = (unsigned long long*)(ws + ADR_OFF);
    float*               preeff  = (float*)(ws + PRE_OFF);
    float*               acomp   = (float*)(ws + ACM_OFF);

    snn_prep<<<(RTOT + 127) / 128, 128, 0, stream>>>(
        V, a, Xbuf, dmap, Xext, outX, outV, outA, flags, preeff);

    snn_compact<<<1, 1024, 0, stream>>>(
        flags, preeff, Wint, Wext, addrTab, acomp, hdr);

    snn_gemm<<<NN / 32, 256, 0, stream>>>(addrTab, acomp, hdr, outV);
}